// WignerKernelFullIterations_49349174231046
// MI455X (gfx1250) — compile-verified
//
#include <hip/hip_runtime.h>

typedef __attribute__((ext_vector_type(2))) float v2f;
typedef __attribute__((ext_vector_type(8))) float v8f;

#define NS 16          // samples per tile (WMMA M dimension)
#define UCOLPAD 121    // odd pitch; col 120 = dummy sink, junk reads stay in-row
#define WPB 4          // waves per block (share one CG pack; 2 WGs / 320KB WGP)

// sum of (2m+1)^2 for m < l : {0,1,10,35}
__device__ __forceinline__ constexpr int TRI(int l) { return (4 * l * l * l - l) / 3; }
// X block element offsets within 84-float per-sample X state
__device__ __forceinline__ constexpr int XO(int l) { return TRI(l); }
// wig block offsets within 168-float per-sample wig state
__device__ __forceinline__ constexpr int WOF(int lam, int sneg) {
  return 2 * TRI(lam) + sneg * (2 * lam + 1) * (2 * lam + 1);
}
// packed-CG pair-region base; region holds all lambda slices contiguously in
// global-q order, in-region offset of C[i,j,q_global] = (qg*d2 + j)*d1 + i
__device__ __forceinline__ constexpr int PB(int l1, int l2) {
  return 16 * ((2 * l1 + 1) * l2 * l2 + 16 * l1 * l1);
}

// One (A-block x X_l2) coupling producing all admissible lambda outputs.
// Step 1 (U = C·A) on V_WMMA_F32_16X16X4_F32, 16 samples on M; kk-outer order
// keeps NCT independent accumulator chains in flight (WMMA ILP at low occupancy).
// Steps 2+3 (Y = U·B, out += Y·C) per-lane, branchless, halves split q by parity.
template<int L1, int L2>
__device__ __forceinline__ void pair_combine(
    const float* __restrict__ srcA, int s1neg,
    const float* __restrict__ srcB,
    float* __restrict__ wnext,
    const float* __restrict__ sC,
    float* __restrict__ sU)
{
  constexpr int D1    = 2 * L1 + 1, D2 = 2 * L2 + 1;
  constexpr int LAMLO = (L1 > L2) ? (L1 - L2) : (L2 - L1);
  constexpr int LAMHI = (L1 + L2 < 3) ? (L1 + L2) : 3;
  constexpr int QG    = (LAMHI + 1) * (LAMHI + 1) - LAMLO * LAMLO;
  constexpr int TOTC  = QG * D2;            // U columns = (qg, j) flattened
  constexpr int NCT   = (TOTC + 15) / 16;   // 16-wide column tiles
  constexpr int NKK   = (D1 + 3) / 4;       // K chunks of 4 for f32 WMMA
  constexpr int PBC   = PB(L1, L2);

  const int lane = threadIdx.x & 31;
  const int n    = lane & 15;
  const int half = lane >> 4;

  // ---- hoisted B-operand (shared CG) fragments: cb = PBC + col*D1 + i ----
  v2f bfr[NCT][NKK];
  #pragma unroll
  for (int ct = 0; ct < NCT; ++ct) {
    const int col  = ct * 16 + n;
    const int colc = (col < TOTC) ? col : 0;      // clamp, load always valid
    #pragma unroll
    for (int kk = 0; kk < NKK; ++kk) {
      v2f b;
      #pragma unroll
      for (int v = 0; v < 2; ++v) {
        const int i  = kk * 4 + v + 2 * half;     // B layout: K = v + 2*(lane>=16)
        const int ic = (i < D1) ? i : 0;
        const float bv = sC[PBC + colc * D1 + ic];
        b[v] = (col < TOTC && i < D1) ? bv : 0.0f; // cndmask, no exec branches
      }
      bfr[ct][kk] = b;
    }
  }

  // ---- hoist per-sample B block into VGPRs (static indices after unroll) ----
  float Breg[D2 * D2];
  #pragma unroll
  for (int j = 0; j < D2; ++j)
    #pragma unroll
    for (int l = 0; l < D2; ++l)
      Breg[j * D2 + l] = srcB[(j * D2 + l) * NS + n];

  #pragma unroll 1
  for (int k = 0; k < D1; ++k) {
    // ---- A-operand fragments A_k[n, i], branchless zero padding ----
    v2f afr[NKK];
    #pragma unroll
    for (int kk = 0; kk < NKK; ++kk) {
      v2f a;
      #pragma unroll
      for (int v = 0; v < 2; ++v) {
        const int i  = kk * 4 + v + 2 * half;     // A layout: K = v + 2*(lane>=16)
        const int ic = (i < D1) ? i : 0;
        const float av = srcA[(ic * D1 + k) * NS + n];
        a[v] = (i < D1) ? av : 0.0f;
      }
      afr[kk] = a;
    }
    // ---- step 1: U_k[n, col] via WMMA; independent chains per column tile ----
    const v8f vz = {0.f, 0.f, 0.f, 0.f, 0.f, 0.f, 0.f, 0.f};
    v8f acc[NCT];
    #pragma unroll
    for (int ct = 0; ct < NCT; ++ct) acc[ct] = vz;
    #pragma unroll
    for (int kk = 0; kk < NKK; ++kk)
      #pragma unroll
      for (int ct = 0; ct < NCT; ++ct)
        acc[ct] = __builtin_amdgcn_wmma_f32_16x16x4_f32(
            false, afr[kk], false, bfr[ct][kk], (short)0, acc[ct], false, false);
    #pragma unroll
    for (int ct = 0; ct < NCT; ++ct) {
      const int col = ct * 16 + n;
      #pragma unroll
      for (int r = 0; r < 8; ++r)               // D layout: M = r + 8*(lane>=16)
        sU[(r + 8 * half) * UCOLPAD + col] = acc[ct][r];
    }
    // ---- steps 2+3: per-lane, branchless; halves split q by parity ----
    int qbase = 0;
    #pragma unroll
    for (int lam = LAMLO; lam <= LAMHI; ++lam) {
      const int dq     = 2 * lam + 1;
      const int qgbase = lam * lam - LAMLO * LAMLO;
      const int dsto   = (s1neg ^ ((L1 + L2 + lam) & 1))
                           ? (2 * TRI(lam) + dq * dq) : (2 * TRI(lam));
      const int par = half ^ (qbase & 1);       // parity alignment for this lam
      #pragma unroll 1
      for (int qh = 0; qh <= lam; ++qh) {       // uniform trip count both halves
        const int q  = 2 * qh + par;            // q == dq possible -> invalid
        const int qg = qbase + q;
        const bool valid = (q < dq);
        float Y[D2];
        #pragma unroll
        for (int l = 0; l < D2; ++l) Y[l] = 0.0f;
        #pragma unroll
        for (int j = 0; j < D2; ++j) {
          const float u = sU[n * UCOLPAD + qg * D2 + j];  // junk-safe if invalid
          #pragma unroll
          for (int l = 0; l < D2; ++l) Y[l] += u * Breg[j * D2 + l];
        }
        #pragma unroll
        for (int p = 0; p < dq; ++p) {
          float a2 = 0.0f;
          #pragma unroll
          for (int l = 0; l < D2; ++l)
            a2 += Y[l] * sC[PBC + ((qgbase + p) * D2 + l) * D1 + k];
          float* dst = valid ? (wnext + (dsto + q * dq + p) * NS + n)
                             : (sU + n * UCOLPAD + 120);   // dummy sink
          *dst += a2;                                      // ptr-select, no branch
        }
      }
      qbase += dq;
    }
  }
}

template<int L1>
__device__ __forceinline__ void combine_row(
    const float* srcA, int s1neg, const float* sX,
    float* wnext, const float* sC, float* sU)
{
  pair_combine<L1, 0>(srcA, s1neg, sX + XO(0) * NS, wnext, sC, sU);
  pair_combine<L1, 1>(srcA, s1neg, sX + XO(1) * NS, wnext, sC, sU);
  pair_combine<L1, 2>(srcA, s1neg, sX + XO(2) * NS, wnext, sC, sU);
  pair_combine<L1, 3>(srcA, s1neg, sX + XO(3) * NS, wnext, sC, sU);
}

// Final lambda_max=0 coupling, scalar [0,0] only: tiny -> pure VALU.
template<int LAM>
__device__ __forceinline__ float final_block(const float* __restrict__ W,
                                             const float* __restrict__ B,
                                             const float* __restrict__ sC, int n)
{
  constexpr int D  = 2 * LAM + 1;
  constexpr int cb = PB(LAM, LAM);  // lam=0 slice starts at region base
  float t[D * D];
  #pragma unroll
  for (int k = 0; k < D; ++k)
    #pragma unroll
    for (int j = 0; j < D; ++j) {
      float s = 0.0f;
      #pragma unroll
      for (int i = 0; i < D; ++i) s += sC[cb + j * D + i] * W[(i * D + k) * NS + n];
      t[k * D + j] = s;
    }
  float r = 0.0f;
  #pragma unroll
  for (int j = 0; j < D; ++j)
    #pragma unroll
    for (int l = 0; l < D; ++l) {
      float s = 0.0f;
      #pragma unroll
      for (int k = 0; k < D; ++k) s += t[k * D + j] * sC[cb + l * D + k];
      r += s * B[(j * D + l) * NS + n];
    }
  return r;
}

__global__ __launch_bounds__(32 * WPB)
void wigner_wmma_kernel(const float* __restrict__ x0, const float* __restrict__ x1,
                        const float* __restrict__ x2, const float* __restrict__ x3,
                        const float* __restrict__ cg, float* __restrict__ out, int N)
{
  __shared__ float sC[4096];                 // packed CG, shared by all 4 waves
  __shared__ float sXs [WPB][84 * NS];       // per-wave X blocks   [elem][sample]
  __shared__ float sW2s[WPB][168 * NS];      // per-wave wig nu=2
  __shared__ float sW3s[WPB][168 * NS];      // per-wave wig nu=3
  __shared__ float sUs [WPB][16 * UCOLPAD];  // per-wave U_k stage

  const int tid  = threadIdx.x;
  const int lane = tid & 31;
  const int wv   = tid >> 5;
  const long long tile = (long long)blockIdx.x * WPB + wv;

  // ---- cooperative CG pack into LDS (once per block, all 128 threads) ----
  for (int l1 = 0; l1 < 4; ++l1)
    for (int l2 = 0; l2 < 4; ++l2) {
      const int d1 = 2 * l1 + 1, d2 = 2 * l2 + 1;
      const int lo = (l1 > l2) ? (l1 - l2) : (l2 - l1);
      const int hi = (l1 + l2 < 3) ? (l1 + l2) : 3;
      for (int lam = lo; lam <= hi; ++lam) {
        const int off = PB(l1, l2) + (lam * lam - lo * lo) * d1 * d2;
        const int sz  = d1 * d2 * (2 * lam + 1);
        const float* src = cg + (size_t)((l1 * 4 + l2) * 4 + lam) * 343;
        for (int idx = tid; idx < sz; idx += 32 * WPB) {
          const int q = idx / (d1 * d2);
          const int r = idx - q * d1 * d2;
          const int j = r / d1;
          const int i = r - j * d1;
          sC[off + idx] = src[i * 49 + j * 7 + q];   // cg[l1,l2,lam,i,j,q]
        }
      }
    }
  __syncthreads();   // the only cross-wave barrier: sC ready

  float* sX  = sXs[wv];
  float* sW2 = sW2s[wv];
  float* sW3 = sW3s[wv];
  float* sU  = sUs[wv];

  // ---- per-wave: load 16-sample X tile (same-wave DS ordering suffices) ----
  #pragma unroll
  for (int l = 0; l < 4; ++l) {
    const float* xsrc = (l == 0) ? x0 : (l == 1) ? x1 : (l == 2) ? x2 : x3;
    const int d = 2 * l + 1, dd = d * d;
    for (int idx = lane; idx < dd * NS; idx += 32) {
      const int nn = idx / dd, e = idx - nn * dd;
      const long long g = tile * NS + nn;
      sX[(XO(l) + e) * NS + nn] = (g < N) ? xsrc[(size_t)g * dd + e] : 0.0f;
    }
  }
  for (int idx = lane; idx < 168 * NS; idx += 32) { sW2[idx] = 0.0f; sW3[idx] = 0.0f; }

  // ---- level 2: X (x) X -> sW2 (lambda <= 3) ----
  combine_row<0>(sX + XO(0) * NS, 0, sX, sW2, sC, sU);
  combine_row<1>(sX + XO(1) * NS, 0, sX, sW2, sC, sU);
  combine_row<2>(sX + XO(2) * NS, 0, sX, sW2, sC, sU);
  combine_row<3>(sX + XO(3) * NS, 0, sX, sW2, sC, sU);

  // ---- level 3: wig2 (x) X -> sW3 (wig2 has no (0,-1) block) ----
  combine_row<0>(sW2 + WOF(0, 0) * NS, 0, sX, sW3, sC, sU);
  #pragma unroll 1
  for (int s = 0; s < 2; ++s) {       // runtime parity loop halves code size
    combine_row<1>(sW2 + (WOF(1, 0) + s * 9)  * NS, s, sX, sW3, sC, sU);
    combine_row<2>(sW2 + (WOF(2, 0) + s * 25) * NS, s, sX, sW3, sC, sU);
    combine_row<3>(sW2 + (WOF(3, 0) + s * 49) * NS, s, sX, sW3, sC, sU);
  }

  // ---- level 4: lambda_max=0, only (0,+1)[0,0]; needs s1=+1 blocks ----
  const int n = lane & 15, half = lane >> 4;
  float part = 0.0f;
  if (half == 0) {
    part += final_block<0>(sW3 + WOF(0, 0) * NS, sX + XO(0) * NS, sC, n);
    part += final_block<1>(sW3 + WOF(1, 0) * NS, sX + XO(1) * NS, sC, n);
  } else {
    part += final_block<2>(sW3 + WOF(2, 0) * NS, sX + XO(2) * NS, sC, n);
    part += final_block<3>(sW3 + WOF(3, 0) * NS, sX + XO(3) * NS, sC, n);
  }
  sU[lane] = part;   // same-wave cross-lane handoff; DS in-order per wave

  const long long g = tile * NS + n;
  if (half == 0 && g < N) {
    float4 o;
    o.x = sX[0 * NS + n];        // x_0_1[n,0,0]
    o.y = sW2[0 * NS + n];       // wig2 (0,+1)[0,0]
    o.z = sW3[0 * NS + n];       // wig3 (0,+1)[0,0]
    o.w = sU[n] + sU[n + 16];    // wig4 (0,+1)[0,0]
    reinterpret_cast<float4*>(out)[g] = o;
  }
}

extern "C" void kernel_launch(void* const* d_in, const int* in_sizes, int n_in,
                              void* d_out, int out_size, void* d_ws, size_t ws_size,
                              hipStream_t stream)
{
  const float* x0 = (const float*)d_in[0];
  const float* x1 = (const float*)d_in[1];
  const float* x2 = (const float*)d_in[2];
  const float* x3 = (const float*)d_in[3];
  const float* cg = (const float*)d_in[4];
  const int N = in_sizes[0];                  // x_0_1 is (N,1,1)
  const int tiles  = (N + NS - 1) / NS;       // N=100000 -> 6250 full tiles
  const int blocks = (tiles + WPB - 1) / WPB; // 4 tiles (waves) per block
  wigner_wmma_kernel<<<blocks, 32 * WPB, 0, stream>>>(x0, x1, x2, x3, cg,
                                                      (float*)d_out, N);
}